// GraphEncoder_91302414778552
// MI455X (gfx1250) — compile-verified
//
#include <hip/hip_runtime.h>

#define IN_DIM   256
#define OUT_DIM  256
#define HEADS    4
#define HID      64
#define NEG_SLOPE 0.2f

typedef float v2f __attribute__((ext_vector_type(2)));
typedef float v8f __attribute__((ext_vector_type(8)));

// ---------------------------------------------------------------------------
// Helpers
// ---------------------------------------------------------------------------
__device__ __forceinline__ float lrelu(float v) {
    return v > 0.0f ? v : NEG_SLOPE * v;
}

// Order-preserving float <-> uint encoding so we can use integer atomicMax.
__device__ __forceinline__ unsigned f2ord(float f) {
    unsigned u = __float_as_uint(f);
    return (u & 0x80000000u) ? ~u : (u | 0x80000000u);
}
__device__ __forceinline__ float ord2f(unsigned u) {
    u = (u & 0x80000000u) ? (u & 0x7fffffffu) : ~u;
    return __uint_as_float(u);
}

__device__ __forceinline__ void edge_pair(const int* __restrict__ ei, int E,
                                          int e, int& s, int& d) {
    if (e < E) { s = ei[e]; d = ei[E + e]; }
    else       { s = e - E; d = e - E; }   // self loop
}

// ---------------------------------------------------------------------------
// Kernel 0: init accumulators (out = 0, max = -inf encoded, denom = 0)
// ---------------------------------------------------------------------------
__global__ void gat_init(float* __restrict__ out, unsigned* __restrict__ mmax,
                         float* __restrict__ denom, long long out_elems, int nh) {
    long long i = (long long)blockIdx.x * blockDim.x + threadIdx.x;
    if (i < out_elems) out[i] = 0.0f;
    if (i < nh) {
        mmax[i]  = 0x007fffffu;  // f2ord(-inf): ~0xff800000
        denom[i] = 0.0f;
    }
}

// ---------------------------------------------------------------------------
// Kernel 1: h = x @ W via V_WMMA_F32_16X16X4_F32.
// Block = 256 threads (8 waves); block computes a 32-row x 256-col strip.
// Wave w owns columns [32w, 32w+32); both 16-row tiles -> 4 accumulators.
// ---------------------------------------------------------------------------
__global__ __launch_bounds__(256) void gat_gemm_wmma(
    const float* __restrict__ x, const float* __restrict__ W,
    float* __restrict__ h, int nrows)
{
    __shared__ float lds_x[32][IN_DIM + 2];  // +2 pad: bank-conflict avoidance
    const int row0 = blockIdx.x * 32;
    const int tid  = threadIdx.x;
    const int wave = tid >> 5;
    const int lane = tid & 31;

    // Stage 32x256 strip of x into LDS (clamp rows at the ragged tail).
    {
        const int r  = tid & 31;
        const int c0 = (tid >> 5) * 32;
        int rr = row0 + r;
        if (rr >= nrows) rr = nrows - 1;
        const float* src = x + (size_t)rr * IN_DIM + c0;
#pragma unroll
        for (int j = 0; j < 32; j += 4) {
            float4 v = *(const float4*)(src + j);
            lds_x[r][c0 + j + 0] = v.x;
            lds_x[r][c0 + j + 1] = v.y;
            lds_x[r][c0 + j + 2] = v.z;
            lds_x[r][c0 + j + 3] = v.w;
        }
    }
    __syncthreads();

    const int m    = lane & 15;            // A-matrix row within tile
    const int kb   = (lane >> 4) << 1;     // lane half selects K pair {0,1}|{2,3}
    const int col  = wave * 32 + (lane & 15);

    v8f acc00 = {}; v8f acc01 = {};
    v8f acc10 = {}; v8f acc11 = {};

    for (int k = 0; k < IN_DIM; k += 4) {
        v2f a0, a1, b0, b1;
        // A: 16x4 f32 tile, lanes 0-15 hold K=k+0/k+1, lanes 16-31 K=k+2/k+3
        a0.x = lds_x[m][k + kb];       a0.y = lds_x[m][k + kb + 1];
        a1.x = lds_x[16 + m][k + kb];  a1.y = lds_x[16 + m][k + kb + 1];
        // B: 4x16 f32 tile from W (row-major [IN_DIM][OUT_DIM])
        const float* wp = W + (size_t)(k + kb) * OUT_DIM + col;
        b0.x = wp[0];   b0.y = wp[OUT_DIM];
        b1.x = wp[16];  b1.y = wp[OUT_DIM + 16];

        acc00 = __builtin_amdgcn_wmma_f32_16x16x4_f32(false, a0, false, b0, (short)0, acc00, false, false);
        acc01 = __builtin_amdgcn_wmma_f32_16x16x4_f32(false, a0, false, b1, (short)0, acc01, false, false);
        acc10 = __builtin_amdgcn_wmma_f32_16x16x4_f32(false, a1, false, b0, (short)0, acc10, false, false);
        acc11 = __builtin_amdgcn_wmma_f32_16x16x4_f32(false, a1, false, b1, (short)0, acc11, false, false);
    }

    // C/D layout: VGPR r -> row (r | lane-half*8), col = lane%16
    const int rbase = (lane >> 4) << 3;
#pragma unroll
    for (int r = 0; r < 8; ++r) {
        int row_lo = row0 + rbase + r;
        int row_hi = row_lo + 16;
        if (row_lo < nrows) {
            h[(size_t)row_lo * OUT_DIM + col]      = acc00[r];
            h[(size_t)row_lo * OUT_DIM + col + 16] = acc01[r];
        }
        if (row_hi < nrows) {
            h[(size_t)row_hi * OUT_DIM + col]      = acc10[r];
            h[(size_t)row_hi * OUT_DIM + col + 16] = acc11[r];
        }
    }
}

// ---------------------------------------------------------------------------
// Kernel 2: per-node attention logits a_src/a_dst [N, HEADS]
// ---------------------------------------------------------------------------
__global__ void gat_logits(const float* __restrict__ h,
                           const float* __restrict__ att_src,
                           const float* __restrict__ att_dst,
                           float* __restrict__ a_src, float* __restrict__ a_dst,
                           int nnodes)
{
    int i = blockIdx.x * blockDim.x + threadIdx.x;  // node*HEADS + head
    if (i >= nnodes * HEADS) return;
    int node = i >> 2, head = i & 3;
    const float* hp = h + (size_t)node * OUT_DIM + head * HID;
    const float* as = att_src + head * HID;
    const float* ad = att_dst + head * HID;
    float s = 0.0f, d = 0.0f;
#pragma unroll 8
    for (int j = 0; j < HID; ++j) {
        float v = hp[j];
        s += v * as[j];
        d += v * ad[j];
    }
    a_src[i] = s;
    a_dst[i] = d;
}

// ---------------------------------------------------------------------------
// Kernel 3: segment max of leaky-relu edge logits (over incoming edges)
// ---------------------------------------------------------------------------
__global__ void gat_edge_max(const int* __restrict__ ei,
                             const float* __restrict__ a_src,
                             const float* __restrict__ a_dst,
                             unsigned* __restrict__ mmax, int E, int nn)
{
    int i = blockIdx.x * blockDim.x + threadIdx.x;
    int total = (E + nn) * HEADS;
    if (i >= total) return;
    int e = i >> 2, hd = i & 3;
    int s, d; edge_pair(ei, E, e, s, d);
    float v = lrelu(a_src[s * HEADS + hd] + a_dst[d * HEADS + hd]);
    atomicMax(&mmax[d * HEADS + hd], f2ord(v));
}

// ---------------------------------------------------------------------------
// Kernel 4: segment sum of exp(e - max)
// ---------------------------------------------------------------------------
__global__ void gat_edge_sum(const int* __restrict__ ei,
                             const float* __restrict__ a_src,
                             const float* __restrict__ a_dst,
                             const unsigned* __restrict__ mmax,
                             float* __restrict__ denom, int E, int nn)
{
    int i = blockIdx.x * blockDim.x + threadIdx.x;
    int total = (E + nn) * HEADS;
    if (i >= total) return;
    int e = i >> 2, hd = i & 3;
    int s, d; edge_pair(ei, E, e, s, d);
    float v = lrelu(a_src[s * HEADS + hd] + a_dst[d * HEADS + hd]);
    float mv = ord2f(mmax[d * HEADS + hd]);
    atomicAdd(&denom[d * HEADS + hd], expf(v - mv));
}

// ---------------------------------------------------------------------------
// Kernel 5: weighted scatter-aggregate. One 256-thread block per edge:
// thread t handles channel t (head = t/64), out[dst,t] += alpha * h[src,t].
// ---------------------------------------------------------------------------
__global__ __launch_bounds__(256) void gat_edge_aggr(
    const int* __restrict__ ei, const float* __restrict__ h,
    const float* __restrict__ a_src, const float* __restrict__ a_dst,
    const unsigned* __restrict__ mmax, const float* __restrict__ denom,
    float* __restrict__ out, int E, int nn)
{
    __shared__ float s_alpha[HEADS];
    int e = blockIdx.x;
    int s, d; edge_pair(ei, E, e, s, d);
    int t = threadIdx.x;
    if (t < HEADS) {
        float v   = lrelu(a_src[s * HEADS + t] + a_dst[d * HEADS + t]);
        float num = expf(v - ord2f(mmax[d * HEADS + t]));
        s_alpha[t] = num / denom[d * HEADS + t];
    }
    __syncthreads();
    float alpha = s_alpha[t >> 6];
    atomicAdd(&out[(size_t)d * OUT_DIM + t], alpha * h[(size_t)s * OUT_DIM + t]);
}

// ---------------------------------------------------------------------------
// Kernel 6: out = relu(out + bias)
// ---------------------------------------------------------------------------
__global__ void gat_finalize(float* __restrict__ out,
                             const float* __restrict__ bias, long long n)
{
    long long i = (long long)blockIdx.x * blockDim.x + threadIdx.x;
    if (i >= n) return;
    float v = out[i] + bias[i & (OUT_DIM - 1)];
    out[i] = v > 0.0f ? v : 0.0f;
}

// ---------------------------------------------------------------------------
// Launch
// ---------------------------------------------------------------------------
extern "C" void kernel_launch(void* const* d_in, const int* in_sizes, int n_in,
                              void* d_out, int out_size, void* d_ws, size_t ws_size,
                              hipStream_t stream)
{
    const float* x    = (const float*)d_in[0];
    const int*   ei   = (const int*)d_in[1];   // jax default x64-off: int32
    const float* W    = (const float*)d_in[2];
    const float* atts = (const float*)d_in[3];
    const float* attd = (const float*)d_in[4];
    const float* bias = (const float*)d_in[5];

    const int N = in_sizes[0] / IN_DIM;
    const int E = in_sizes[1] / 2;
    float* out = (float*)d_out;

    // Workspace layout (all f32/u32, 4-byte aligned):
    float*    h     = (float*)d_ws;                         // N * 256
    float*    a_src = h + (size_t)N * OUT_DIM;              // N * 4
    float*    a_dst = a_src + (size_t)N * HEADS;            // N * 4
    unsigned* mmax  = (unsigned*)(a_dst + (size_t)N * HEADS); // N * 4
    float*    denom = (float*)(mmax + (size_t)N * HEADS);   // N * 4

    const long long out_elems = (long long)N * OUT_DIM;

    gat_init<<<(int)((out_elems + 255) / 256), 256, 0, stream>>>(
        out, mmax, denom, out_elems, N * HEADS);

    gat_gemm_wmma<<<(N + 31) / 32, 256, 0, stream>>>(x, W, h, N);

    gat_logits<<<(N * HEADS + 255) / 256, 256, 0, stream>>>(
        h, atts, attd, a_src, a_dst, N);

    const int TE = E + N;  // edges + self loops
    gat_edge_max<<<(TE * HEADS + 255) / 256, 256, 0, stream>>>(
        ei, a_src, a_dst, mmax, E, N);

    gat_edge_sum<<<(TE * HEADS + 255) / 256, 256, 0, stream>>>(
        ei, a_src, a_dst, mmax, denom, E, N);

    gat_edge_aggr<<<TE, 256, 0, stream>>>(
        ei, h, a_src, a_dst, mmax, denom, out, E, N);

    gat_finalize<<<(int)((out_elems + 255) / 256), 256, 0, stream>>>(
        out, bias, out_elems);
}